// DivergenceFreeKernel_37907381354774
// MI455X (gfx1250) — compile-verified
//
#include <hip/hip_runtime.h>
#include <hip/hip_bf16.h>
#include <math.h>

// Problem constants (fixed by the reference)
#define NN 2048
#define MM 2048
#define DD 16

// Workspace layout (floats)
#define WS_XL   0                      // N*16  : X / ls
#define WS_AW   (WS_XL  + NN*DD)       // N*16  : X / ls^2
#define WS_X2L  (WS_AW  + NN*DD)       // M*16  : X2 / ls
#define WS_BW   (WS_X2L + MM*DD)       // M*16  : X2 / ls^2
#define WS_XS   (WS_BW  + MM*DD)       // N     : ||Xl_n||^2
#define WS_AS   (WS_XS  + NN)          // N     : ||A_n||^2
#define WS_X2S  (WS_AS  + NN)          // M     : ||X2l_m||^2
#define WS_BS   (WS_X2S + MM)          // M     : ||B_m||^2
#define WS_S    (WS_BS  + MM)          // scalar: sum_d 1/ls^2
#define WS_VAR  (WS_S + 1)             // scalar: softplus(uv)

typedef float v2f __attribute__((ext_vector_type(2)));
typedef float v8f __attribute__((ext_vector_type(8)));

__device__ __forceinline__ float softplus_f(float x) {
  return (x > 20.0f) ? x : log1pf(expf(x));
}

// Prologue: scale rows, compute row norms, compute scalars.
__global__ __launch_bounds__(256) void dfk_prep(const float* __restrict__ X,
                                                const float* __restrict__ X2,
                                                const float* __restrict__ uls,
                                                const float* __restrict__ uv,
                                                float* __restrict__ ws) {
  int r = blockIdx.x * 256 + threadIdx.x;   // 0 .. 4095 (rows of X then X2)

  float inv[DD], l2[DD];
  float S = 0.0f;
#pragma unroll
  for (int d = 0; d < DD; ++d) {
    float ls = softplus_f(uls[d]);
    float iv = 1.0f / ls;
    inv[d] = iv;
    l2[d] = iv * iv;
    S += iv * iv;
  }

  if (r == 0) {
    ws[WS_S]   = S;
    ws[WS_VAR] = softplus_f(uv[0]);
  }

  if (r < NN + MM) {
    bool first = (r < NN);
    int  rr    = first ? r : (r - NN);
    const float* src = (first ? X  : X2) + rr * DD;
    float* dl  = ws + (first ? WS_XL : WS_X2L) + rr * DD;
    float* dw  = ws + (first ? WS_AW : WS_BW)  + rr * DD;
    float* dn1 = ws + (first ? WS_XS : WS_X2S) + rr;
    float* dn2 = ws + (first ? WS_AS : WS_BS)  + rr;

    float ns = 0.0f, na = 0.0f;
#pragma unroll
    for (int d = 0; d < DD; ++d) {
      float v  = src[d];
      float xl = v * inv[d];
      float aw = v * l2[d];
      dl[d] = xl;
      dw[d] = aw;
      ns += xl * xl;
      na += aw * aw;
    }
    *dn1 = ns;
    *dn2 = na;
  }
}

// Main fused kernel: one wave32 computes one 16x16 output tile via
// two chains of V_WMMA_F32_16X16X4_F32 (K=16 -> 4 chained WMMAs each).
__global__ __launch_bounds__(256) void dfk_main(const float* __restrict__ ws,
                                                float* __restrict__ out) {
  const float* Xl  = ws + WS_XL;
  const float* Aw  = ws + WS_AW;
  const float* X2l = ws + WS_X2L;
  const float* Bw  = ws + WS_BW;
  const float* Xs  = ws + WS_XS;
  const float* Asn = ws + WS_AS;
  const float* X2s = ws + WS_X2S;
  const float* Bsn = ws + WS_BS;
  const float  S   = ws[WS_S];
  const float  var = ws[WS_VAR];

  const int lane = threadIdx.x & 31;
  const int wave = threadIdx.x >> 5;
  const int h    = lane >> 4;   // half-wave: selects K-subrange / M-subrange
  const int lr   = lane & 15;   // row within tile (A-M index and B/C/D N index)

  const int n0 = (blockIdx.y * 8 + wave) * 16;  // 8 waves/block walk N-tiles
  const int m0 = blockIdx.x * 16;               // block shares the M-tile

  const float* xrow = Xl  + (n0 + lr) * DD;
  const float* yrow = X2l + (m0 + lr) * DD;
  const float* arow = Aw  + (n0 + lr) * DD;
  const float* brow = Bw  + (m0 + lr) * DD;

  v8f c_sq = {};
  v8f c_w  = {};
#pragma unroll
  for (int s = 0; s < 4; ++s) {
    const int k = 4 * s + 2 * h;  // A/B VGPR layout for 16x16x4 f32 (2 VGPRs)
    v2f a  = *(const v2f*)(xrow + k);
    v2f b  = *(const v2f*)(yrow + k);
    c_sq = __builtin_amdgcn_wmma_f32_16x16x4_f32(
        /*neg_a=*/false, a, /*neg_b=*/false, b,
        /*c_mod=*/(short)0, c_sq, /*reuse_a=*/false, /*reuse_b=*/false);
    v2f aa = *(const v2f*)(arow + k);
    v2f bb = *(const v2f*)(brow + k);
    c_w = __builtin_amdgcn_wmma_f32_16x16x4_f32(
        /*neg_a=*/false, aa, /*neg_b=*/false, bb,
        /*c_mod=*/(short)0, c_w, /*reuse_a=*/false, /*reuse_b=*/false);
  }

  const float x2s = X2s[m0 + lr];
  const float bs  = Bsn[m0 + lr];
  const int   m   = m0 + lr;

#pragma unroll
  for (int v = 0; v < 8; ++v) {
    const int   n   = n0 + 8 * h + v;            // C/D layout: M = v + 8*h
    const float sq  = Xs[n]  + x2s - 2.0f * c_sq[v];
    const float wsq = Asn[n] + bs  - 2.0f * c_w[v];
    const float val = var * expf(-0.5f * sq) * (wsq + ((float)(DD - 1) - sq) * S);
    out[n * MM + m] = val;
  }
}

extern "C" void kernel_launch(void* const* d_in, const int* in_sizes, int n_in,
                              void* d_out, int out_size, void* d_ws, size_t ws_size,
                              hipStream_t stream) {
  const float* X   = (const float*)d_in[0];
  const float* X2  = (const float*)d_in[1];
  const float* uls = (const float*)d_in[2];
  const float* uv  = (const float*)d_in[3];
  float* out = (float*)d_out;
  float* ws  = (float*)d_ws;

  // Prologue: 4096 rows, one thread per row.
  dfk_prep<<<dim3((NN + MM) / 256), dim3(256), 0, stream>>>(X, X2, uls, uv, ws);

  // Main: 128 x 128 tiles of 16x16; 8 waves per block (each wave = 1 tile).
  dfk_main<<<dim3(MM / 16, NN / 16 / 8), dim3(256), 0, stream>>>(ws, out);
}